// ExpHashEncoder_90623809945986
// MI455X (gfx1250) — compile-verified
//
#include <hip/hip_runtime.h>
#include <hip/hip_bf16.h>

// ---------------- problem constants (from reference) ----------------
#define NPTS        524288
#define NFRAMES     4
#define BASIS_NUM   8
#define NLEVELS     16
#define MAXP        524288            // 2^19
#define TOTALP      7131219LL         // sum of per-level params
#define T2          14262438LL        // TOTALP * 2 (floats per basis row)

typedef __attribute__((ext_vector_type(2))) float v2f;
typedef __attribute__((ext_vector_type(8))) float v8f;

// basis row b (0 = embeddings_mean, 1..7 = embeddings[b-1]) element j
__device__ __forceinline__ float basis_val(const float* __restrict__ mean,
                                           const float* __restrict__ emb,
                                           int b, long long j) {
  return (b == 0) ? mean[j] : emb[(size_t)(b - 1) * T2 + (size_t)j];
}

// ---------------------------------------------------------------
// Kernel 1: per-level basis blend  cur[f, j] = sum_b exp[f,b]*basis[b, off2+j]
// D = A x B with V_WMMA_F32_16X16X4_F32 (two K=4 steps, K=8 total):
//   A (16x4): rows = frame f (only rows 0..3 valid), cols = basis index k
//   B (4x16): rows = basis index k, cols = 16 consecutive j values
//   D (16x16): row f, col j_local  -> dense coalesced stores
// A VGPR layout (ISA): V0 = {K0 lanes0-15 | K2 lanes16-31}, V1 = {K1 | K3}
// B VGPR layout (sym): V0 = {K0 | K2}, V1 = {K1 | K3}; lane-within-half = N
// D VGPR layout: Vv lanes0-15 -> (M=v, N=lane); lanes16-31 -> (M=v+8, N-16)
// ---------------------------------------------------------------
__global__ void blend_wmma_kernel(const float* __restrict__ mean,
                                  const float* __restrict__ emb,
                                  const float* __restrict__ expw,   // [4][8]
                                  float* __restrict__ cur,          // [4][nElems]
                                  long long off2, int nElems) {
  const int lane = threadIdx.x & 31;
  const int wave = threadIdx.x >> 5;
  const int tile = blockIdx.x * (blockDim.x >> 5) + wave;
  const int tiles = (nElems + 15) >> 4;
  if (tile >= tiles) return;                 // wave-uniform exit, EXEC stays all-1

  const int half = lane >> 4;                // 0: lanes 0-15, 1: lanes 16-31
  const int m    = lane & 15;
  const int n    = tile * 16 + m;
  const int nc   = (n < nElems) ? n : (nElems - 1);  // clamp loads: no divergence
  const long long j = off2 + (long long)nc;

  // A matrix: exp (rows = frame, cols = k).  Row index fm = m&3 replica for m>=4
  // (those D rows are never stored).
  const int fm = m & 3;
  v2f a0, a1;
  a0.x = expw[fm * 8 + half * 2 + 0];        // K0 | K2
  a0.y = expw[fm * 8 + half * 2 + 1];        // K1 | K3
  a1.x = expw[fm * 8 + half * 2 + 4];        // second K-chunk (b = 4..7)
  a1.y = expw[fm * 8 + half * 2 + 5];

  // B matrix: basis tile (rows = k, cols = j_local)
  v2f b0, b1;
  b0.x = basis_val(mean, emb, half * 2 + 0, j);
  b0.y = basis_val(mean, emb, half * 2 + 1, j);
  b1.x = basis_val(mean, emb, half * 2 + 4, j);
  b1.y = basis_val(mean, emb, half * 2 + 5, j);

  v8f acc = {};
  acc = __builtin_amdgcn_wmma_f32_16x16x4_f32(false, a0, false, b0,
                                              (short)0, acc, false, false);
  acc = __builtin_amdgcn_wmma_f32_16x16x4_f32(false, a1, false, b1,
                                              (short)0, acc, false, false);

  // D row f=v lives in acc[v], lanes 0-15, col = lane.  Frames 0..3 only.
  if (half == 0 && n < nElems) {
#pragma unroll
    for (int v = 0; v < NFRAMES; ++v)
      cur[(size_t)v * (size_t)nElems + (size_t)n] = acc[v];
  }
}

// ---------------------------------------------------------------
// Kernel 2: per-level trilinear hash-grid gather.
//   FUSED=false: read blended cur[frame][idx][c] from workspace (L2-resident)
//   FUSED=true : fallback, dot 8 basis rows with exp[frame] per corner
// ---------------------------------------------------------------
template <bool FUSED>
__global__ void gather_kernel(const float* __restrict__ inputs,   // [N][3]
                              const int* __restrict__ rays,       // [N]
                              const float* __restrict__ cur,      // [4][hsize][2]
                              const float* __restrict__ mean,
                              const float* __restrict__ emb,
                              const float* __restrict__ expw,
                              float* __restrict__ out,            // [N][32]
                              int level, float scale, int res,
                              int hsize, long long off, int dense) {
  int p = blockIdx.x * blockDim.x + threadIdx.x;
  if (p >= NPTS) return;

  float x = inputs[3 * p + 0];
  float y = inputs[3 * p + 1];
  float z = inputs[3 * p + 2];

  // pos = x*scale + 0.5 with exact separate mul/add (no FMA contraction:
  // a 1-ulp change can flip the grid cell)
  float px = __fadd_rn(__fmul_rn(x, scale), 0.5f);
  float py = __fadd_rn(__fmul_rn(y, scale), 0.5f);
  float pz = __fadd_rn(__fmul_rn(z, scale), 0.5f);
  float gx = floorf(px), gy = floorf(py), gz = floorf(pz);
  float fx = __fsub_rn(px, gx), fy = __fsub_rn(py, gy), fz = __fsub_rn(pz, gz);
  int ix = (int)gx, iy = (int)gy, iz = (int)gz;
  float ox = __fsub_rn(1.0f, fx), oy = __fsub_rn(1.0f, fy), oz = __fsub_rn(1.0f, fz);

  int frame = rays[p] >> 10;

  float e[BASIS_NUM];
  if constexpr (FUSED) {
#pragma unroll
    for (int b = 0; b < BASIS_NUM; ++b) e[b] = expw[frame * 8 + b];
  }

  const int stride = res + 1;
  const unsigned mask = (unsigned)(hsize - 1);   // hash levels: hsize = 2^19
  float s0 = 0.0f, s1 = 0.0f;

#pragma unroll
  for (int c = 0; c < 8; ++c) {
    const int cx = (c >> 2) & 1, cy = (c >> 1) & 1, cz = c & 1;
    unsigned idx;
    if (dense) {
      idx = (unsigned)((ix + cx) + (iy + cy) * stride +
                       (iz + cz) * stride * stride);
    } else {
      unsigned h = (unsigned)(ix + cx) * 1u ^
                   (unsigned)(iy + cy) * 2654435761u ^
                   (unsigned)(iz + cz) * 805459861u;
      idx = h & mask;
    }
    float w = ((cx ? fx : ox) * (cy ? fy : oy)) * (cz ? fz : oz);

    float v0, v1;
    if constexpr (!FUSED) {
      const float2 vv = *(const float2*)(cur +
          ((size_t)frame * (size_t)hsize + (size_t)idx) * 2);
      v0 = vv.x; v1 = vv.y;
    } else {
      size_t jj = ((size_t)off + (size_t)idx) * 2;
      v0 = e[0] * mean[jj];
      v1 = e[0] * mean[jj + 1];
#pragma unroll
      for (int b = 1; b < BASIS_NUM; ++b) {
        const float* r = emb + (size_t)(b - 1) * T2 + jj;
        v0 += e[b] * r[0];
        v1 += e[b] * r[1];
      }
    }
    s0 += w * v0;
    s1 += w * v1;
  }

  out[(size_t)p * 32 + 2 * level + 0] = s0;
  out[(size_t)p * 32 + 2 * level + 1] = s1;
}

// ---------------------------------------------------------------
extern "C" void kernel_launch(void* const* d_in, const int* in_sizes, int n_in,
                              void* d_out, int out_size, void* d_ws, size_t ws_size,
                              hipStream_t stream) {
  (void)in_sizes; (void)n_in; (void)out_size;
  const float* inputs = (const float*)d_in[0];
  const float* expw   = (const float*)d_in[1];
  const float* mean   = (const float*)d_in[2];
  const float* emb    = (const float*)d_in[3];
  const int*   rays   = (const int*)d_in[4];
  float* out = (float*)d_out;
  float* cur = (float*)d_ws;

  // per-level workspace: 4 frames * hsize * 2 floats, max 16 MiB
  const size_t need = (size_t)NFRAMES * (size_t)MAXP * 2u * sizeof(float);
  const bool useWS = (cur != nullptr) && (ws_size >= need);

  const int gatherBlocks = (NPTS + 255) / 256;

  long long off = 0;
  for (int l = 0; l < NLEVELS; ++l) {
    const long long resll = 16LL << l;              // 16 * 2^l exactly
    const int res = (int)resll;
    const long long dense_sz = (resll + 1) * (resll + 1) * (resll + 1);
    const long long hs = dense_sz < (long long)MAXP ? dense_sz : (long long)MAXP;
    const int hsize = (int)hs;
    const int dense = (dense_sz <= hs) ? 1 : 0;
    const float scale = (float)(resll - 1);         // BASE_RES*2^l - 1, exact
    const int nElems = hsize * 2;

    if (useWS) {
      const int tiles = (nElems + 15) / 16;
      const int blocks = (tiles + 7) / 8;           // 8 waves / 256-thread block
      blend_wmma_kernel<<<blocks, 256, 0, stream>>>(mean, emb, expw, cur,
                                                    off * 2, nElems);
      gather_kernel<false><<<gatherBlocks, 256, 0, stream>>>(
          inputs, rays, cur, mean, emb, expw, out,
          l, scale, res, hsize, off, dense);
    } else {
      gather_kernel<true><<<gatherBlocks, 256, 0, stream>>>(
          inputs, rays, nullptr, mean, emb, expw, out,
          l, scale, res, hsize, off, dense);
    }
    off += hs;
  }
}